// Conditional_Diffusion_Model_6700148981808
// MI455X (gfx1250) — compile-verified
//
#include <hip/hip_runtime.h>
#include <hip/hip_bf16.h>
#include <math.h>

// ---------------------------------------------------------------------------
// Conditional diffusion model loss, CDNA5 (gfx1250) wave32 + WMMA f32 16x16x4
//
// Dead-code analysis of the reference:
//   * error_pro is zeroed -> pro_x/pro_h/Wp/bp/eps_h_pro never matter; only
//     pro_idx contributes (joint segment-mean of x_noise).
//   * xh_mol is a broadcast segment mean, so
//       eps_hat = C[seg] - sigma*(eps @ Wz),  C = alpha*(mean@Wz)+t*Wt+bm
//       d       = eps + sigma*(eps @ Wz) - C[seg]
//   * t==0 branch collapses to segment-level math (sigma=0 => z_t = mean).
// ---------------------------------------------------------------------------

typedef __attribute__((ext_vector_type(2))) float v2f;
typedef __attribute__((ext_vector_type(8))) float v8f;

#define XD 3
#define NA 64
#define DF 67           /* XD + NA */
#define KP 68           /* K padded to multiple of 4 */
#define TDIV 1000.0f

// ---------------------------------------------------------------------------
// Kernel A: one block per segment. Segment sums/means, alpha/sigma, C vector.
// ---------------------------------------------------------------------------
__global__ __launch_bounds__(256) void seg_prep(
    const float* __restrict__ mol_x, const float* __restrict__ mol_h,
    const float* __restrict__ x_noise,
    const float* __restrict__ Wm, const float* __restrict__ bm,
    const float* __restrict__ t,
    const int* __restrict__ mol_idx, const int* __restrict__ pro_idx,
    int Nm, int Np,
    float* __restrict__ sumh, float* __restrict__ meanv,
    float* __restrict__ Cseg, float* __restrict__ jm,
    float* __restrict__ sig, float* __restrict__ cnt,
    float* __restrict__ err, float* __restrict__ err3)
{
  const int b   = blockIdx.x;
  const int tid = threadIdx.x;
  __shared__ int   bnds[4];
  __shared__ float smean[DF + 1];
  __shared__ float sjm[3];
  __shared__ float red[256];

  if (tid < 4) {
    const int* arr = (tid < 2) ? mol_idx : pro_idx;
    int n   = (tid < 2) ? Nm : Np;
    int key = b + (tid & 1);
    int lo = 0, hi = n;
    while (lo < hi) { int mid = (lo + hi) >> 1; if (arr[mid] < key) lo = mid + 1; else hi = mid; }
    bnds[tid] = lo;
  }
  __syncthreads();
  const int mlo = bnds[0], mhi = bnds[1], plo = bnds[2], phi = bnds[3];
  const float cntm  = (float)(mhi - mlo);
  const float cntmD = fmaxf(cntm, 1.0f);
  const float cntjD = fmaxf(cntm + (float)(phi - plo), 1.0f);

  // ---- mol_h segment sums: col = tid&63, 4 row-groups ----
  {
    const int col = tid & 63, grp = tid >> 6;
    float acc = 0.f;
    for (int r = mlo + grp; r < mhi; r += 4)
      acc += mol_h[(size_t)r * NA + col];
    red[tid] = acc;
    __syncthreads();
    if (grp == 0) {
      float s = red[col] + red[64 + col] + red[128 + col] + red[192 + col];
      sumh[(size_t)b * NA + col] = s;                 // raw sum (for t0 path)
      smean[XD + col] = s / (4.0f * cntmD);           // normalized (h/NV[1]) mean
    }
    __syncthreads();
  }
  // ---- mol_x segment means (NV[0] = 1) ----
  for (int col = 0; col < 3; ++col) {
    float a = 0.f;
    for (int r = mlo + tid; r < mhi; r += 256) a += mol_x[(size_t)r * 3 + col];
    red[tid] = a; __syncthreads();
    for (int s = 128; s > 0; s >>= 1) { if (tid < s) red[tid] += red[tid + s]; __syncthreads(); }
    if (tid == 0) smean[col] = red[0] / cntmD;
    __syncthreads();
  }
  // ---- joint x_noise mean over mol rows + pro rows of this segment ----
  for (int col = 0; col < 3; ++col) {
    float a = 0.f;
    for (int r = mlo + tid; r < mhi; r += 256) a += x_noise[(size_t)r * 3 + col];
    for (int r = plo + tid; r < phi; r += 256) a += x_noise[(size_t)(Nm + r) * 3 + col];
    red[tid] = a; __syncthreads();
    for (int s = 128; s > 0; s >>= 1) { if (tid < s) red[tid] += red[tid + s]; __syncthreads(); }
    if (tid == 0) sjm[col] = red[0] / cntjD;
    __syncthreads();
  }

  const float tb    = t[b];
  const float tt    = tb / TDIV;
  const float alpha = 1.0f - tt * tt;
  const float sigma = sqrtf(fmaxf(0.0f, 1.0f - alpha * alpha));

  // ---- C[b][n] = bm[n] + t*Wm[67][n] + alpha * sum_k mean[k]*Wm[k][n] ----
  if (tid < DF) {
    float s2 = 0.f;
    for (int k = 0; k < DF; ++k) s2 += smean[k] * Wm[(size_t)k * DF + tid];
    Cseg[(size_t)b * DF + tid]  = bm[tid] + tb * Wm[(size_t)DF * DF + tid] + alpha * s2;
    meanv[(size_t)b * DF + tid] = smean[tid];
  }
  if (tid == 0) {
    jm[b * 3 + 0] = sjm[0]; jm[b * 3 + 1] = sjm[1]; jm[b * 3 + 2] = sjm[2];
    sig[b]  = sigma;
    cnt[b]  = cntm;
    err[b]  = 0.0f;
    err3[b] = 0.0f;
  }
}

// ---------------------------------------------------------------------------
// Kernel B: per-node GEMM + error via V_WMMA_F32_16X16X4_F32.
// One wave per 16-row tile; K padded 67->68 (17 steps of 4); N in 5 tiles.
// K-order inside LDS: [h0..h63, x0..x2, 0]  (keeps float4 stores aligned);
// Wm rows reordered to match when loading into LDS.
// ---------------------------------------------------------------------------
__global__ __launch_bounds__(256) void node_err(
    const float* __restrict__ x_noise, const float* __restrict__ eps_h,
    const int* __restrict__ mol_idx, const float* __restrict__ Wm,
    int Nm,
    const float* __restrict__ Cseg, const float* __restrict__ jm,
    const float* __restrict__ sig,
    float* __restrict__ err, float* __restrict__ err3)
{
  __shared__ float Wl[KP][80];       // reordered weights, N padded to 80
  __shared__ float eA[8][16][72];    // per-wave epsilon tile, stride 72 (288B)
  __shared__ int   sSeg[8][16];

  const int tid = threadIdx.x;
  for (int i = tid; i < KP * 80; i += 256) {
    int k = i / 80, n = i % 80;
    float v = 0.f;
    if (n < DF && k < DF) {
      int kr = (k < NA) ? (XD + k) : (k - NA);   // h rows first, then x rows
      v = Wm[(size_t)kr * DF + n];
    }
    Wl[k][n] = v;
  }
  __syncthreads();

  const int wave = tid >> 5, lane = tid & 31;
  const int ln = lane & 15, hh = lane >> 4;
  const int ntiles = (Nm + 15) >> 4;

  for (int tile = blockIdx.x * 8 + wave; tile < ntiles; tile += gridDim.x * 8) {
    const int base = tile << 4;
    __builtin_amdgcn_wave_barrier();   // previous-iteration LDS reads done (in-order DS)

    // ---- stage 16 rows: 2 lanes per row, 32 h-floats each via float4 ----
    {
      const int r    = lane >> 1;
      const int half = lane & 1;
      int row  = base + r;
      int rowc = (row < Nm) ? row : (Nm - 1);
      const float4* hp = (const float4*)(eps_h + (size_t)rowc * NA + half * 32);
      float4* dst = (float4*)(&eA[wave][r][half * 32]);
#pragma unroll
      for (int j = 0; j < 8; ++j) dst[j] = hp[j];
      // prefetch next tile's slice of eps_h
      __builtin_prefetch(eps_h + (size_t)((rowc + 16 < Nm) ? rowc + 16 : rowc) * NA + half * 32, 0, 1);
      if (half == 0) {
        int seg = mol_idx[rowc];
        sSeg[wave][r] = seg;
        const float* xp = x_noise + (size_t)rowc * 3;
        eA[wave][r][64] = xp[0] - jm[seg * 3 + 0];
        eA[wave][r][65] = xp[1] - jm[seg * 3 + 1];
        eA[wave][r][66] = xp[2] - jm[seg * 3 + 2];
        eA[wave][r][67] = 0.0f;                      // K pad
      }
    }
    __builtin_amdgcn_wave_barrier();

    // ---- A fragments in registers: lane = row ln, K = 4*ki + 2*hh (+1) ----
    v2f afrag[17];
#pragma unroll
    for (int ki = 0; ki < 17; ++ki)
      afrag[ki] = *(const v2f*)(&eA[wave][ln][ki * 4 + 2 * hh]);

    const int segLo = sSeg[wave][0];
    const int segHi = sSeg[wave][15];
    const float sigLo = sig[segLo];
    const float sigHi = sig[segHi];
    float aLo = 0.f, aHi = 0.f, aLo3 = 0.f, aHi3 = 0.f;

#pragma unroll
    for (int n0 = 0; n0 < 5; ++n0) {
      v8f acc = {0.f, 0.f, 0.f, 0.f, 0.f, 0.f, 0.f, 0.f};
#pragma unroll
      for (int ki = 0; ki < 17; ++ki) {
        const int k0 = ki * 4 + 2 * hh;
        v2f bfrag;
        bfrag[0] = Wl[k0][n0 * 16 + ln];
        bfrag[1] = Wl[k0 + 1][n0 * 16 + ln];
        acc = __builtin_amdgcn_wmma_f32_16x16x4_f32(
            false, afrag[ki], false, bfrag, (short)0, acc, false, false);
      }
      const int  N      = n0 * 16 + ln;
      const bool nvalid = (N < DF);
#pragma unroll
      for (int j = 0; j < 8; ++j) {
        const int M   = j + 8 * hh;
        const int row = base + M;
        if (nvalid && row < Nm) {
          const int   seg = sSeg[wave][M];
          const float u   = acc[j];
          const float e   = (N < XD) ? eA[wave][M][64 + N] : eA[wave][M][N - XD];
          const float sgm = (seg == segLo) ? sigLo : sigHi;
          const float c   = Cseg[(size_t)seg * DF + N];
          const float d   = e + sgm * u - c;
          const float q   = d * d;
          if (seg == segLo) { aLo += q; if (N < XD) aLo3 += q; }
          else              { aHi += q; if (N < XD) aHi3 += q; }
        }
      }
    }
    // ---- wave reduction (wave32) + per-segment atomics ----
#pragma unroll
    for (int off = 16; off > 0; off >>= 1) {
      aLo  += __shfl_xor(aLo,  off, 32);
      aHi  += __shfl_xor(aHi,  off, 32);
      aLo3 += __shfl_xor(aLo3, off, 32);
      aHi3 += __shfl_xor(aHi3, off, 32);
    }
    if (lane == 0) {
      atomicAdd(&err[segLo],  aLo);
      atomicAdd(&err3[segLo], aLo3);
      if (segHi != segLo) { atomicAdd(&err[segHi], aHi); atomicAdd(&err3[segHi], aHi3); }
    }
  }
}

// ---------------------------------------------------------------------------
// Kernel C: per-segment loss assembly (incl. segment-level t==0 branch) and
// mean over B -> scalar output.
// ---------------------------------------------------------------------------
__global__ __launch_bounds__(256) void finalize(
    const float* __restrict__ t,
    const float* __restrict__ sumh, const float* __restrict__ meanv,
    const float* __restrict__ sig,  const float* __restrict__ cnt,
    const float* __restrict__ err,  const float* __restrict__ err3,
    int Nm, int Bseg, float* __restrict__ out)
{
  __shared__ float red[256];
  const int tid = threadIdx.x;
  float acc = 0.f;
  for (int b = tid; b < Bseg; b += 256) {
    const float tb = t[b];
    float lb;
    if (tb != 0.0f) {
      // loss_t = 0.5 * error_mol / ((Nm + NA) * mol_size); error_pro == 0
      lb = 0.5f * err[b] / ((float)(Nm + NA) * fmaxf(cnt[b], 1.0f));
    } else {
      // t==0: sigma=0, alpha=1 -> z_t == segment mean; segment-level t0 terms
      const float s = sig[b] * 4.0f;   // sigma_0 unnormalized (== 0 here)
      const float* mv = meanv + (size_t)b * DF;
      const float* sh = sumh  + (size_t)b * NA;
      float lp[NA];
      float mx = -__builtin_inff();
      for (int ch = 0; ch < NA; ++ch) {
        float c  = mv[XD + ch] * 4.0f - 1.0f;
        float lu = logf(0.5f * (1.0f + erff((c + 0.5f) / s) * 0.70710678118f)
                      - 0.5f * (1.0f + erff((c - 0.5f) / s) * 0.70710678118f) + 1e-10f);
        lp[ch] = lu;
        mx = fmaxf(mx, lu);
      }
      float se = 0.f;
      for (int ch = 0; ch < NA; ++ch) se += expf(lp[ch] - mx);
      const float lse = mx + logf(se);
      float lh = 0.f;                     // sum_rows logp . mol_h = logp . (sum_h/4)
      for (int ch = 0; ch < NA; ++ch) lh += (lp[ch] - lse) * (sh[ch] * 0.25f);
      const float loss_h = -lh;
      const float loss_x = 0.5f * err3[b] / (float)Nm * cnt[b];
      lb = loss_x + loss_h;
    }
    acc += lb;
  }
  red[tid] = acc; __syncthreads();
  for (int s = 128; s > 0; s >>= 1) { if (tid < s) red[tid] += red[tid + s]; __syncthreads(); }
  if (tid == 0) out[0] = red[0] / (float)Bseg;
}

// ---------------------------------------------------------------------------
extern "C" void kernel_launch(void* const* d_in, const int* in_sizes, int n_in,
                              void* d_out, int out_size, void* d_ws, size_t ws_size,
                              hipStream_t stream) {
  const float* mol_x   = (const float*)d_in[0];
  const float* mol_h   = (const float*)d_in[1];
  // d_in[2] pro_x, d_in[3] pro_h : dead (error_pro zeroed in reference)
  const float* Wm      = (const float*)d_in[4];
  const float* bm      = (const float*)d_in[5];
  // d_in[6] Wp, d_in[7] bp : dead
  const float* t       = (const float*)d_in[8];
  const float* x_noise = (const float*)d_in[9];
  const float* eps_h   = (const float*)d_in[10];
  // d_in[11] eps_h_pro : dead
  const int*   mol_idx = (const int*)d_in[12];
  const int*   pro_idx = (const int*)d_in[13];

  const int Nm   = in_sizes[0] / 3;
  const int Np   = in_sizes[2] / 3;
  const int Bseg = in_sizes[8];

  // workspace layout (floats)
  float* w      = (float*)d_ws;
  float* ws_sumh = w;                      // B*64
  float* ws_mean = ws_sumh + (size_t)Bseg * NA;   // B*67
  float* ws_C    = ws_mean + (size_t)Bseg * DF;   // B*67
  float* ws_jm   = ws_C    + (size_t)Bseg * DF;   // B*3
  float* ws_sig  = ws_jm   + (size_t)Bseg * 3;    // B
  float* ws_cnt  = ws_sig  + (size_t)Bseg;        // B
  float* ws_err  = ws_cnt  + (size_t)Bseg;        // B
  float* ws_err3 = ws_err  + (size_t)Bseg;        // B

  seg_prep<<<Bseg, 256, 0, stream>>>(mol_x, mol_h, x_noise, Wm, bm, t,
                                     mol_idx, pro_idx, Nm, Np,
                                     ws_sumh, ws_mean, ws_C, ws_jm,
                                     ws_sig, ws_cnt, ws_err, ws_err3);

  const int ntiles = (Nm + 15) >> 4;
  const int nblk   = (ntiles + 7) >> 3;   // one 16-row tile per wave (8 waves/block)
  node_err<<<nblk, 256, 0, stream>>>(x_noise, eps_h, mol_idx, Wm, Nm,
                                     ws_C, ws_jm, ws_sig, ws_err, ws_err3);

  finalize<<<1, 256, 0, stream>>>(t, ws_sumh, ws_mean, ws_sig, ws_cnt,
                                  ws_err, ws_err3, Nm, Bseg, (float*)d_out);
}